// Criterion_36945308680559
// MI455X (gfx1250) — compile-verified
//
#include <hip/hip_runtime.h>

typedef __attribute__((ext_vector_type(2))) float v2f;
typedef __attribute__((ext_vector_type(8))) float v8f;

#define N_CLOTH      16384
#define N_OBS_VERTS  8192
#define N_OBS_FACES  16384
#define EPS_C        0.001f

// k_nn_wmma staging geometry
#define CHUNK_FACES  256                        // faces per LDS chunk (4 KB)
#define N_CHUNKS     (N_OBS_FACES / CHUNK_FACES)  // 64
#define TILES_PER_CHUNK (CHUNK_FACES / 16)      // 16

// ---------------------------------------------------------------------------
// Workspace layout (bytes):
//   fb   : float4[N_OBS_FACES]  prev face center (x,y,z) + bias = -0.5*|c|^2
//   fp   : float4[N_OBS_FACES]  current face center (x,y,z,0)
//   fn   : float4[N_OBS_FACES]  current unit face normal (x,y,z,0)
//   nn   : int   [N_CLOTH]      argmin face index per cloth vertex
//   part : float [64]           per-block partial sums
// ---------------------------------------------------------------------------
#define WS_FB_OFF   0
#define WS_FP_OFF   (WS_FB_OFF + N_OBS_FACES * 16)
#define WS_FN_OFF   (WS_FP_OFF + N_OBS_FACES * 16)
#define WS_NN_OFF   (WS_FN_OFF + N_OBS_FACES * 16)
#define WS_PART_OFF (WS_NN_OFF + N_CLOTH * 4)

__device__ __forceinline__ float3 ld3(const float* __restrict__ p, int i) {
    return make_float3(p[3 * i + 0], p[3 * i + 1], p[3 * i + 2]);
}

// ---------------------------------------------------------------------------
// Kernel 1: per-face precompute (centers, bias, unit normal)
// ---------------------------------------------------------------------------
__global__ void __launch_bounds__(256)
k_face_precompute(const float* __restrict__ obs_pos,
                  const float* __restrict__ obs_prev,
                  const int*   __restrict__ faces,
                  float4* __restrict__ fb,
                  float4* __restrict__ fp,
                  float4* __restrict__ fn) {
    int f = blockIdx.x * blockDim.x + threadIdx.x;
    if (f >= N_OBS_FACES) return;

    int i0 = faces[3 * f + 0];
    int i1 = faces[3 * f + 1];
    int i2 = faces[3 * f + 2];

    // previous face center + NN bias
    float3 pa = ld3(obs_prev, i0), pb = ld3(obs_prev, i1), pc = ld3(obs_prev, i2);
    float cx = (pa.x + pb.x + pc.x) * (1.0f / 3.0f);
    float cy = (pa.y + pb.y + pc.y) * (1.0f / 3.0f);
    float cz = (pa.z + pb.z + pc.z) * (1.0f / 3.0f);
    float bias = -0.5f * (cx * cx + cy * cy + cz * cz);
    fb[f] = make_float4(cx, cy, cz, bias);

    // current face center
    float3 a = ld3(obs_pos, i0), b = ld3(obs_pos, i1), c = ld3(obs_pos, i2);
    fp[f] = make_float4((a.x + b.x + c.x) * (1.0f / 3.0f),
                        (a.y + b.y + c.y) * (1.0f / 3.0f),
                        (a.z + b.z + c.z) * (1.0f / 3.0f), 0.0f);

    // current unit normal: cross(b-a, c-a) / max(|.|, 1e-12)
    float e1x = b.x - a.x, e1y = b.y - a.y, e1z = b.z - a.z;
    float e2x = c.x - a.x, e2y = c.y - a.y, e2z = c.z - a.z;
    float nx = e1y * e2z - e1z * e2y;
    float ny = e1z * e2x - e1x * e2z;
    float nz = e1x * e2y - e1y * e2x;
    float len = sqrtf(nx * nx + ny * ny + nz * nz);
    float inv = 1.0f / fmaxf(len, 1e-12f);
    fn[f] = make_float4(nx * inv, ny * inv, nz * inv, 0.0f);
}

// ---------------------------------------------------------------------------
// Kernel 2: WMMA brute-force 1-NN with async global->LDS double buffering.
//
// One wave32 owns a 16-row cloth block. Per 16-face tile, one
// v_wmma_f32_16x16x4_f32 computes score[m][n] = a_m . b_n - 0.5*|b_n|^2
// (the bias rides in the K=4 pad slot: A.k3 = 1, B.k3 = bias).
// argmin(d2) == argmax(score).
//
// The face table is streamed into LDS in 4 KB chunks with
// GLOBAL_LOAD_ASYNC_TO_LDS_B128 (ASYNCcnt), double buffered so the DMA of
// chunk c+1 overlaps the 16 WMMAs of chunk c; 8 waves share each chunk.
// ---------------------------------------------------------------------------
__global__ void __launch_bounds__(256)
k_nn_wmma(const float* __restrict__ cloth_prev,
          const float4* __restrict__ fb,
          int* __restrict__ nn) {
    __shared__ float4 fbuf[2][CHUNK_FACES];     // 2 x 4 KB

    const int tid  = threadIdx.x;
    const int lane = tid & 31;
    const int sub  = lane & 15;
    const bool lo  = lane < 16;
    const int wave = blockIdx.x * (blockDim.x >> 5) + (tid >> 5);
    const int m0   = wave * 16;

    // A tile (loop invariant): lanes 0-15 -> (x_m, y_m); lanes 16-31 -> (z_m, 1)
    const int m = m0 + sub;
    float ax = cloth_prev[3 * m + 0];
    float ay = cloth_prev[3 * m + 1];
    float az = cloth_prev[3 * m + 2];
    v2f A;
    A.x = lo ? ax : az;
    A.y = lo ? ay : 1.0f;

    float best[8];
    int   bidx[8];
#pragma unroll
    for (int v = 0; v < 8; ++v) { best[v] = -3.0e38f; bidx[v] = 0; }

    // ---- async stage of chunk 0 (each thread moves one float4) ----
    {
        unsigned ldsa = (unsigned)(uintptr_t)&fbuf[0][tid];
        unsigned long long ga = (unsigned long long)(uintptr_t)(fb + tid);
        asm volatile("global_load_async_to_lds_b128 %0, %1, off"
                     :: "v"(ldsa), "v"(ga) : "memory");
    }

    for (int c = 0; c < N_CHUNKS; ++c) {
        if (c + 1 < N_CHUNKS) {
            // stage chunk c+1 into the other buffer while we compute chunk c
            unsigned ldsa = (unsigned)(uintptr_t)&fbuf[(c + 1) & 1][tid];
            unsigned long long ga =
                (unsigned long long)(uintptr_t)(fb + (c + 1) * CHUNK_FACES + tid);
            asm volatile("global_load_async_to_lds_b128 %0, %1, off"
                         :: "v"(ldsa), "v"(ga) : "memory");
            asm volatile("s_wait_asynccnt 0x1" ::: "memory"); // chunk c landed
        } else {
            asm volatile("s_wait_asynccnt 0x0" ::: "memory");
        }
        __syncthreads();                         // chunk c visible to all waves

        const float4* __restrict__ buf = fbuf[c & 1];
        const int nbase = c * CHUNK_FACES + sub;

#pragma unroll 4
        for (int tl = 0; tl < TILES_PER_CHUNK; ++tl) {
            const int n = nbase + tl * 16;
            float4 f4 = buf[tl * 16 + sub];      // ds_load_b128 (broadcast pairs)

            // B tile: lanes 0-15 -> (x_n, y_n); lanes 16-31 -> (z_n, bias_n)
            v2f B;
            B.x = lo ? f4.x : f4.z;
            B.y = lo ? f4.y : f4.w;

            v8f C = {};
            v8f D = __builtin_amdgcn_wmma_f32_16x16x4_f32(
                /*neg_a=*/false, A, /*neg_b=*/false, B,
                /*c_mod=*/(short)0, C, /*reuse_a=*/false, /*reuse_b=*/false);

            // D layout: VGPR v, lanes 0-15 -> row v, lanes 16-31 -> row v+8
#pragma unroll
            for (int v = 0; v < 8; ++v) {
                if (D[v] > best[v]) { best[v] = D[v]; bidx[v] = n; }
            }
        }
        __syncthreads();   // all waves done reading buf before it is re-staged
    }

    // Butterfly reduction across the 16 lanes of each half-wave
    // (xor offsets 1..8 never cross the lane-16 boundary in wave32).
#pragma unroll
    for (int v = 0; v < 8; ++v) {
        float bv = best[v];
        int   bi = bidx[v];
#pragma unroll
        for (int off = 8; off > 0; off >>= 1) {
            float ov = __shfl_xor(bv, off, 32);
            int   oi = __shfl_xor(bi, off, 32);
            if (ov > bv || (ov == bv && oi < bi)) { bv = ov; bi = oi; }
        }
        if (sub == 0) nn[m0 + v + (lo ? 0 : 8)] = bi;  // lane 0: rows 0-7, lane 16: rows 8-15
    }
}

// ---------------------------------------------------------------------------
// Kernel 3: per-vertex penalty + per-block reduction (deterministic)
// ---------------------------------------------------------------------------
__global__ void __launch_bounds__(256)
k_penalty(const float* __restrict__ cloth_pred,
          const float4* __restrict__ fp,
          const float4* __restrict__ fn,
          const int* __restrict__ nn,
          float* __restrict__ part) {
    __shared__ float red[256];
    int i = blockIdx.x * 256 + threadIdx.x;
    float val = 0.0f;
    if (i < N_CLOTH) {
        int k = nn[i];
        float4 p = fp[k];
        float4 q = fn[k];
        float dx = cloth_pred[3 * i + 0] - p.x;
        float dy = cloth_pred[3 * i + 1] - p.y;
        float dz = cloth_pred[3 * i + 2] - p.z;
        float dist = dx * q.x + dy * q.y + dz * q.z;
        float t = fmaxf(EPS_C - dist, 0.0f);
        val = t * t * t;
    }
    red[threadIdx.x] = val;
    __syncthreads();
#pragma unroll
    for (int s = 128; s > 0; s >>= 1) {
        if (threadIdx.x < s) red[threadIdx.x] += red[threadIdx.x + s];
        __syncthreads();
    }
    if (threadIdx.x == 0) part[blockIdx.x] = red[0];
}

// ---------------------------------------------------------------------------
// Kernel 4: final reduction of 64 partials * ramped weight(iteration)
// ---------------------------------------------------------------------------
__global__ void k_finalize(const float* __restrict__ part,
                           const int* __restrict__ iteration,
                           float* __restrict__ out) {
    __shared__ float red[64];
    red[threadIdx.x] = part[threadIdx.x];
    __syncthreads();
#pragma unroll
    for (int s = 32; s > 0; s >>= 1) {
        if (threadIdx.x < s) red[threadIdx.x] += red[threadIdx.x + s];
        __syncthreads();
    }
    if (threadIdx.x == 0) {
        int it = iteration[0] - 50000;
        if (it < 0) it = 0;
        float progress = fminf((float)it * (1.0f / 100000.0f), 1.0f);
        float weight = 1.0f + (5000.0f - 1.0f) * progress;
        out[0] = red[0] * weight;
    }
}

// ---------------------------------------------------------------------------
extern "C" void kernel_launch(void* const* d_in, const int* in_sizes, int n_in,
                              void* d_out, int out_size, void* d_ws, size_t ws_size,
                              hipStream_t stream) {
    (void)in_sizes; (void)n_in; (void)out_size; (void)ws_size;

    const float* obs_pos    = (const float*)d_in[0];
    const float* obs_prev   = (const float*)d_in[1];
    const int*   faces      = (const int*)d_in[2];
    const float* cloth_prev = (const float*)d_in[3];
    const float* cloth_pred = (const float*)d_in[4];
    const int*   iteration  = (const int*)d_in[5];

    char* ws = (char*)d_ws;
    float4* fb  = (float4*)(ws + WS_FB_OFF);
    float4* fp  = (float4*)(ws + WS_FP_OFF);
    float4* fn  = (float4*)(ws + WS_FN_OFF);
    int*    nn  = (int*)(ws + WS_NN_OFF);
    float*  prt = (float*)(ws + WS_PART_OFF);

    k_face_precompute<<<N_OBS_FACES / 256, 256, 0, stream>>>(
        obs_pos, obs_prev, faces, fb, fp, fn);

    // 1024 waves (16 cloth rows each) = 128 blocks x 8 waves
    k_nn_wmma<<<(N_CLOTH / 16) / 8, 256, 0, stream>>>(cloth_prev, fb, nn);

    k_penalty<<<N_CLOTH / 256, 256, 0, stream>>>(cloth_pred, fp, fn, nn, prt);

    k_finalize<<<1, 64, 0, stream>>>(prt, iteration, (float*)d_out);
}